// RGATConv_17575006175419
// MI455X (gfx1250) — compile-verified
//
#include <hip/hip_runtime.h>

#define V_N   80000
#define E_N   640000
#define DIN   128
#define DOUT  128
#define NREL  500
#define NRELP 512           // rel rows padded to multiple of 64
#define NFREQ 65            // rfft bins for n=128
#define WFP   144           // padded spectral width (2*65=130 -> 144, mult of 16)
#define EPB   64            // edges (or nodes) per block in message kernels

typedef __attribute__((ext_vector_type(2))) float v2f;
typedef __attribute__((ext_vector_type(8))) float v8f;

// ---------------------------------------------------------------------------
// Aligned fp32 WMMA GEMM: C = A(MxK) @ B(KxN).  M % (16*RT) == 0, N % 16 == 0,
// K % 4 == 0, lda/ldb even.  Each wave computes RT stacked 16x16 tiles so the
// B fragment is loaded once and reused RT times.  No bounds checks anywhere.
// ---------------------------------------------------------------------------
template <int RT>
__global__ __launch_bounds__(256) void k_gemm_aligned(
    const float* __restrict__ A, const float* __restrict__ B,
    float* __restrict__ C, int M, int N, int K, int lda, int ldb, int ldc) {
  int tilesN = N >> 4;
  int nTasks = (M / (16 * RT)) * tilesN;
  int wave = blockIdx.x * (blockDim.x >> 5) + (threadIdx.x >> 5);
  int nWaves = gridDim.x * (blockDim.x >> 5);
  int lane = threadIdx.x & 31;
  int hw = lane >> 4, lr = lane & 15;
  const v8f vzero = {0, 0, 0, 0, 0, 0, 0, 0};
  for (int task = wave; task < nTasks; task += nWaves) {
    int row0 = (task / tilesN) * 16 * RT;
    int col = (task % tilesN) * 16 + lr;
    v8f acc[RT];
#pragma unroll
    for (int rt = 0; rt < RT; ++rt) acc[rt] = vzero;
    for (int k0 = 0; k0 < K; k0 += 4) {
      int k = k0 + 2 * hw;
      v2f b;
      b.x = B[(size_t)k * ldb + col];
      b.y = B[(size_t)(k + 1) * ldb + col];
#pragma unroll
      for (int rt = 0; rt < RT; ++rt) {
        v2f a = *(const v2f*)&A[(size_t)(row0 + rt * 16 + lr) * lda + k];
        acc[rt] = __builtin_amdgcn_wmma_f32_16x16x4_f32(false, a, false, b,
                                                        (short)0, acc[rt],
                                                        false, false);
      }
    }
#pragma unroll
    for (int rt = 0; rt < RT; ++rt)
#pragma unroll
      for (int v = 0; v < 8; ++v)
        C[(size_t)(row0 + rt * 16 + v + 8 * hw) * ldc + col] = acc[rt][v];
  }
}

// ---------------- build padded DFT matrices ---------------------------------
__global__ void k_build_dft(float* __restrict__ Dfwd, float* __restrict__ Dinv) {
  int idx = blockIdx.x * blockDim.x + threadIdx.x;
  const float TWO_PI_N = 6.283185307179586f / 128.0f;
  if (idx < DIN * WFP) {           // Dfwd[j][c]: c=2f -> cos, 2f+1 -> -sin
    int j = idx / WFP, c = idx % WFP;
    float v = 0.f;
    if (c < 2 * NFREQ) {
      int f = c >> 1;
      float th = TWO_PI_N * (float)(f * j);
      v = (c & 1) ? -__sinf(th) : __cosf(th);
    }
    Dfwd[idx] = v;
  }
  if (idx < WFP * DIN) {           // Dinv[row][k], rows >= 130 zero
    int rrow = idx / DIN, k = idx % DIN;
    float v = 0.f;
    if (rrow < 2 * NFREQ) {
      int f = rrow >> 1;
      float wgt = (f == 0 || f == NFREQ - 1) ? 1.0f : 2.0f;
      float th = TWO_PI_N * (float)(f * k);
      v = ((rrow & 1) ? -wgt * __sinf(th) : wgt * __cosf(th)) * (1.0f / 128.0f);
    }
    Dinv[idx] = v;
  }
}

// ---------------- pack rel / loop_rel into zero-padded buffers --------------
__global__ void k_pack(const float* __restrict__ rel,
                       const float* __restrict__ loop_rel,
                       float* __restrict__ relpad, float* __restrict__ looppad) {
  int i = blockIdx.x * blockDim.x + threadIdx.x;
  if (i < NRELP * DIN) relpad[i] = (i < NREL * DIN) ? rel[i] : 0.f;
  if (i < 16 * DIN) looppad[i] = (i < DIN) ? loop_rel[i] : 0.f;
}

// ---------------- ordered-uint mapping for float atomic max -----------------
__device__ __forceinline__ unsigned f2ord(float f) {
  unsigned u = __float_as_uint(f);
  return u ^ ((u & 0x80000000u) ? 0xFFFFFFFFu : 0x80000000u);
}
__device__ __forceinline__ float ord2f(unsigned k) {
  unsigned u = k ^ ((k & 0x80000000u) ? 0x80000000u : 0xFFFFFFFFu);
  return __uint_as_float(u);
}

__global__ void k_init(float* h, float* esum, float* bnsum, float* bnsq,
                       unsigned* emax) {
  size_t i = (size_t)blockIdx.x * blockDim.x + threadIdx.x;
  if (i < (size_t)V_N * DOUT) h[i] = 0.f;
  if (i < V_N) { esum[i] = 0.f; emax[i] = 0u; }  // key 0 decodes NaN -> 0
  if (i < DOUT) { bnsum[i] = 0.f; bnsq[i] = 0.f; }
}

// ---------------- edge attention: leaky_relu of split-weight dots -----------
__global__ __launch_bounds__(256) void k_attn(
    const float* __restrict__ x, const float* __restrict__ rel,
    const float* __restrict__ attn_w, const int* __restrict__ src,
    const int* __restrict__ dst, const int* __restrict__ etype,
    float* __restrict__ e_att, unsigned* __restrict__ emax) {
  int e = blockIdx.x * (blockDim.x >> 5) + (threadIdx.x >> 5);
  if (e >= E_N) return;
  int lane = threadIdx.x & 31;
  int s = src[e], d = dst[e], t = etype[e];
  float acc = 0.f;
#pragma unroll
  for (int i = 0; i < 4; ++i) {
    int c = lane + 32 * i;
    acc += x[(size_t)s * DIN + c] * attn_w[c] +
           rel[(size_t)t * DIN + c] * attn_w[DIN + c] +
           x[(size_t)d * DIN + c] * attn_w[2 * DIN + c];
  }
  for (int off = 16; off; off >>= 1) acc += __shfl_down(acc, off, 32);
  if (lane == 0) {
    float v = acc > 0.f ? acc : 0.01f * acc;
    e_att[e] = v;
    atomicMax(&emax[d], f2ord(v));
  }
}

__global__ void k_expsum(const float* __restrict__ e_att,
                         const int* __restrict__ dst,
                         const unsigned* __restrict__ emax,
                         float* __restrict__ ee, float* __restrict__ esum) {
  int e = blockIdx.x * blockDim.x + threadIdx.x;
  if (e >= E_N) return;
  int d = dst[e];
  float m = ord2f(emax[d]);
  if (!(m > -3.0e38f && m < 3.0e38f)) m = 0.f;
  float v = __expf(e_att[e] - m);
  ee[e] = v;
  atomicAdd(&esum[d], v);
}

// ---------------------------------------------------------------------------
// Per-edge spectral message, 64 edges/block (E and E/2 are multiples of 64 so
// no validity checks and each block is entirely in one half):
//   z = conj(FX[src]) * FR[type];  msg = z @ P_half (WMMA, B reused 4x);
//   softmax(dim=1) * edge_norm * alpha;  atomic scatter into h[dst].
// ---------------------------------------------------------------------------
__global__ __launch_bounds__(256) void k_edge_msg(
    const float* __restrict__ FX, const float* __restrict__ FR,
    const float* __restrict__ Pin, const float* __restrict__ Pout,
    const int* __restrict__ src, const int* __restrict__ dst,
    const int* __restrict__ etype, const float* __restrict__ ee,
    const float* __restrict__ esum, const float* __restrict__ edge_norm,
    float* __restrict__ h) {
  __shared__ __align__(16) float zt[EPB][WFP + 4];  // stride 148: 8B aligned
  __shared__ float msg[EPB][DOUT];
  __shared__ float red[EPB][4];
  int e0 = blockIdx.x * EPB;
  int tid = threadIdx.x;
  int r = tid >> 2, c = tid & 3;          // 4 threads per edge row
  int e = e0 + r;
  {
    const float* fxr = FX + (size_t)src[e] * WFP;
    const float* frr = FR + (size_t)etype[e] * WFP;
    for (int f = c; f < WFP / 2; f += 4) {
      float zre = 0.f, zim = 0.f;
      if (f < NFREQ) {
        float ar = fxr[2 * f], ai = fxr[2 * f + 1];
        float br = frr[2 * f], bi = frr[2 * f + 1];
        zre = ar * br + ai * bi;          // conj(a)*b
        zim = ar * bi - ai * br;
      }
      zt[r][2 * f] = zre;
      zt[r][2 * f + 1] = zim;
    }
  }
  __syncthreads();
  const float* P = (e0 < E_N / 2) ? Pin : Pout;
  int lane = tid & 31;
  int hw = lane >> 4, lr = lane & 15;
  int col = (tid >> 5) * 16 + lr;
  const v8f vzero = {0, 0, 0, 0, 0, 0, 0, 0};
  v8f acc[4];
#pragma unroll
  for (int rt = 0; rt < 4; ++rt) acc[rt] = vzero;
  for (int k0 = 0; k0 < WFP; k0 += 4) {
    int k = k0 + 2 * hw;
    v2f b;
    b.x = P[(size_t)k * DOUT + col];
    b.y = P[(size_t)(k + 1) * DOUT + col];
#pragma unroll
    for (int rt = 0; rt < 4; ++rt) {
      v2f a = *(const v2f*)&zt[rt * 16 + lr][k];
      acc[rt] = __builtin_amdgcn_wmma_f32_16x16x4_f32(false, a, false, b,
                                                      (short)0, acc[rt],
                                                      false, false);
    }
  }
#pragma unroll
  for (int rt = 0; rt < 4; ++rt)
#pragma unroll
    for (int v = 0; v < 8; ++v) msg[rt * 16 + v + 8 * hw][col] = acc[rt][v];
  __syncthreads();
  // row softmax over 128 columns (4 threads per row, 32 cols each)
  float pm = -3.0e38f;
  for (int j = c; j < DOUT; j += 4) pm = fmaxf(pm, msg[r][j]);
  red[r][c] = pm;
  __syncthreads();
  if (c == 0) {
    float m2 = fmaxf(fmaxf(red[r][0], red[r][1]), fmaxf(red[r][2], red[r][3]));
    red[r][0] = m2;
  }
  __syncthreads();
  float rowmax = red[r][0];
  float ps = 0.f;
  for (int j = c; j < DOUT; j += 4) ps += __expf(msg[r][j] - rowmax);
  __syncthreads();
  red[r][c] = ps;
  __syncthreads();
  if (c == 0) red[r][0] = red[r][0] + red[r][1] + red[r][2] + red[r][3];
  __syncthreads();
  float sumexp = red[r][0];
  int d = dst[e];
  float es = esum[d];
  float alpha = ee[e] / (es > 0.f ? es : 1.f);
  float coef = alpha * edge_norm[e] / sumexp;
  for (int j = c; j < DOUT; j += 4)
    atomicAdd(&h[(size_t)d * DOUT + j], __expf(msg[r][j] - rowmax) * coef);
}

// ---------------------------------------------------------------------------
// Self-loop branch, 64 nodes/block (V % 64 == 0): h += (conj(FX)*FLR) @ Ploop
// with h preloaded into the WMMA C operand; unique writer per element.
// ---------------------------------------------------------------------------
__global__ __launch_bounds__(256) void k_loop(
    const float* __restrict__ FX, const float* __restrict__ FLR,
    const float* __restrict__ Ploop, float* __restrict__ h) {
  __shared__ __align__(16) float zt[EPB][WFP + 4];
  int n0 = blockIdx.x * EPB;
  int tid = threadIdx.x;
  int r = tid >> 2, c = tid & 3;
  {
    const float* fxr = FX + (size_t)(n0 + r) * WFP;
    for (int f = c; f < WFP / 2; f += 4) {
      float zre = 0.f, zim = 0.f;
      if (f < NFREQ) {
        float ar = fxr[2 * f], ai = fxr[2 * f + 1];
        float br = FLR[2 * f], bi = FLR[2 * f + 1];
        zre = ar * br + ai * bi;
        zim = ar * bi - ai * br;
      }
      zt[r][2 * f] = zre;
      zt[r][2 * f + 1] = zim;
    }
  }
  __syncthreads();
  int lane = tid & 31;
  int hw = lane >> 4, lr = lane & 15;
  int col = (tid >> 5) * 16 + lr;
  v8f acc[4];
#pragma unroll
  for (int rt = 0; rt < 4; ++rt)
#pragma unroll
    for (int v = 0; v < 8; ++v)
      acc[rt][v] = h[(size_t)(n0 + rt * 16 + v + 8 * hw) * DOUT + col];
  for (int k0 = 0; k0 < WFP; k0 += 4) {
    int k = k0 + 2 * hw;
    v2f b;
    b.x = Ploop[(size_t)k * DOUT + col];
    b.y = Ploop[(size_t)(k + 1) * DOUT + col];
#pragma unroll
    for (int rt = 0; rt < 4; ++rt) {
      v2f a = *(const v2f*)&zt[rt * 16 + lr][k];
      acc[rt] = __builtin_amdgcn_wmma_f32_16x16x4_f32(false, a, false, b,
                                                      (short)0, acc[rt],
                                                      false, false);
    }
  }
#pragma unroll
  for (int rt = 0; rt < 4; ++rt)
#pragma unroll
    for (int v = 0; v < 8; ++v)
      h[(size_t)(n0 + rt * 16 + v + 8 * hw) * DOUT + col] = acc[rt][v];
}

// ---------- out_pre = h/3 + bias; per-block column partial sums for BN ------
__global__ __launch_bounds__(128) void k_finalize1(
    const float* __restrict__ h, const float* __restrict__ bias,
    float* __restrict__ out, float* __restrict__ bnsum, float* __restrict__ bnsq) {
  int c = threadIdx.x;
  int r0 = blockIdx.x * 64;
  float s = 0.f, sq = 0.f;
  float bc = bias[c];
  for (int i = 0; i < 64; ++i) {
    int rI = r0 + i;
    float v = h[(size_t)rI * DOUT + c] * (1.0f / 3.0f) + bc;
    out[(size_t)rI * DOUT + c] = v;
    s += v;
    sq += v * v;
  }
  atomicAdd(&bnsum[c], s);
  atomicAdd(&bnsq[c], sq);
}

__global__ void k_bnapply(float* __restrict__ out,
                          const float* __restrict__ bnsum,
                          const float* __restrict__ bnsq) {
  size_t i = (size_t)blockIdx.x * blockDim.x + threadIdx.x;
  if (i >= (size_t)V_N * DOUT) return;
  int c = (int)(i % DOUT);
  float mean = bnsum[c] * (1.0f / V_N);
  float var = bnsq[c] * (1.0f / V_N) - mean * mean;
  out[i] = (out[i] - mean) * rsqrtf(var + 1e-5f);
}

// ---------------------------------------------------------------------------
template <int RT>
static inline void launch_gemm(const float* A, const float* B, float* C,
                               int M, int N, int K, int lda, int ldb, int ldc,
                               hipStream_t stream) {
  int tasks = (M / (16 * RT)) * (N / 16);
  int blocks = (tasks + 7) / 8;   // 8 waves per 256-thread block
  k_gemm_aligned<RT><<<blocks, 256, 0, stream>>>(A, B, C, M, N, K, lda, ldb, ldc);
}

extern "C" void kernel_launch(void* const* d_in, const int* in_sizes, int n_in,
                              void* d_out, int out_size, void* d_ws, size_t ws_size,
                              hipStream_t stream) {
  (void)in_sizes; (void)n_in; (void)out_size; (void)ws_size;
  const float* x        = (const float*)d_in[0];
  const float* rel      = (const float*)d_in[4];
  const float* edge_nrm = (const float*)d_in[5];
  const float* in_w     = (const float*)d_in[6];
  const float* out_w    = (const float*)d_in[7];
  const float* loop_w   = (const float*)d_in[8];
  const float* w_rel    = (const float*)d_in[9];
  const float* loop_rel = (const float*)d_in[10];
  const float* attn_w   = (const float*)d_in[11];
  const float* bias     = (const float*)d_in[12];
  const int* src        = (const int*)d_in[13];
  const int* dst        = (const int*)d_in[14];
  const int* etype      = (const int*)d_in[15];

  float* out    = (float*)d_out;                 // [V][DOUT]
  float* relout = out + (size_t)V_N * DOUT;      // [NREL][DOUT]

  float* w = (float*)d_ws;
  size_t off = 0;
  auto alloc = [&](size_t n) {
    float* p = w + off;
    off += (n + 63) & ~(size_t)63;               // 256B-aligned slices
    return p;
  };
  float* Dfwd   = alloc((size_t)DIN * WFP);
  float* Dinv   = alloc((size_t)WFP * DIN);
  float* Pin    = alloc((size_t)WFP * DOUT);
  float* Pout   = alloc((size_t)WFP * DOUT);
  float* Ploop  = alloc((size_t)WFP * DOUT);
  float* relpad = alloc((size_t)NRELP * DIN);
  float* looppad= alloc((size_t)16 * DIN);
  float* FR     = alloc((size_t)NRELP * WFP);
  float* FLR    = alloc((size_t)16 * WFP);
  float* FX     = alloc((size_t)V_N * WFP);
  float* e_att  = alloc((size_t)E_N);
  float* ee     = alloc((size_t)E_N);
  float* esum   = alloc((size_t)V_N);
  unsigned* emax = (unsigned*)alloc((size_t)V_N);
  float* hbuf   = alloc((size_t)V_N * DOUT);
  float* bnsum  = alloc(DOUT);
  float* bnsq   = alloc(DOUT);
  float* relout_pad = alloc((size_t)NRELP * DOUT);

  k_init<<<((size_t)V_N * DOUT + 255) / 256, 256, 0, stream>>>(hbuf, esum, bnsum, bnsq, emax);
  k_build_dft<<<(WFP * DIN + 255) / 256, 256, 0, stream>>>(Dfwd, Dinv);
  k_pack<<<(NRELP * DIN + 255) / 256, 256, 0, stream>>>(rel, loop_rel, relpad, looppad);

  // spectra via DFT-as-GEMM (all WMMA, fully aligned)
  launch_gemm<4>(x, Dfwd, FX, V_N, WFP, DIN, DIN, WFP, WFP, stream);
  launch_gemm<4>(relpad, Dfwd, FR, NRELP, WFP, DIN, DIN, WFP, WFP, stream);
  launch_gemm<1>(looppad, Dfwd, FLR, 16, WFP, DIN, DIN, WFP, WFP, stream);
  // fused projection matrices P = Dinv @ W
  launch_gemm<1>(Dinv, in_w, Pin, WFP, DOUT, DIN, DIN, DOUT, DOUT, stream);
  launch_gemm<1>(Dinv, out_w, Pout, WFP, DOUT, DIN, DIN, DOUT, DOUT, stream);
  launch_gemm<1>(Dinv, loop_w, Ploop, WFP, DOUT, DIN, DIN, DOUT, DOUT, stream);
  // second output: rel @ w_rel (padded, then slice-copy into d_out)
  launch_gemm<4>(relpad, w_rel, relout_pad, NRELP, DOUT, DIN, DIN, DOUT, DOUT, stream);
  hipMemcpyAsync(relout, relout_pad, (size_t)NREL * DOUT * sizeof(float),
                 hipMemcpyDeviceToDevice, stream);

  // attention + segment softmax normalizers
  k_attn<<<E_N / 8, 256, 0, stream>>>(x, rel, attn_w, src, dst, etype, e_att, emax);
  k_expsum<<<(E_N + 255) / 256, 256, 0, stream>>>(e_att, dst, emax, ee, esum);

  // per-edge spectral message GEMM + softmax + scatter (64 edges per block)
  k_edge_msg<<<E_N / EPB, 256, 0, stream>>>(FX, FR, Pin, Pout, src, dst, etype,
                                            ee, esum, edge_nrm, hbuf);
  // self-loop GEMM accumulating into hbuf (64 nodes per block)
  k_loop<<<V_N / EPB, 256, 0, stream>>>(FX, FLR, Ploop, hbuf);

  // bias + /3 + BatchNorm (two passes)
  k_finalize1<<<V_N / 64, 128, 0, stream>>>(hbuf, bias, out, bnsum, bnsq);
  k_bnapply<<<((size_t)V_N * DOUT + 255) / 256, 256, 0, stream>>>(out, bnsum, bnsq);
}